// UpscalingMethod_35450660061237
// MI455X (gfx1250) — compile-verified
//
#include <hip/hip_runtime.h>

// Nearest-neighbor 4x block upscale: [B,C,H,W] f32 -> [B,C,4H,4W] f32
// B=4, C=3, H=W=512 (fixed by the reference setup), scale_factor=4.
//
// Pure bandwidth problem: 12 MB read + 192 MB write => ~8.8 us floor @ 23.3 TB/s.
// Strategy: thread t owns 4 consecutive input pixels (one b128 load), emits
// 16 fully-coalesced b128 non-temporal stores (4 output rows x 4 quads).

typedef float v4f __attribute__((ext_vector_type(4)));

__global__ __launch_bounds__(256) void
UpscalingMethod_35450660061237_kernel(const float* __restrict__ in,
                                      float* __restrict__ out,
                                      int n_quads) {
    const int t = blockIdx.x * blockDim.x + threadIdx.x;
    if (t >= n_quads) return;

    // t indexes quads of 4 input pixels along x. W=512 -> 128 quads/row.
    const int ix4 = (t & 127) << 2;        // first input x of this quad
    const int iy  = (t >> 7) & 511;        // input y
    const int bc  = t >> 16;               // fused batch*channel index

    // One wide NT load of 4 adjacent source pixels (each read exactly once).
    const v4f v = __builtin_nontemporal_load(
        reinterpret_cast<const v4f*>(in) + t);

    // Output base (in floats): bc plane of 2048x2048, row 4*iy, col 4*ix4.
    const int obase = bc * (2048 * 2048) + (iy << 2) * 2048 + (ix4 << 2);
    v4f* __restrict__ orow = reinterpret_cast<v4f*>(out + obase);

    const v4f s0 = { v.x, v.x, v.x, v.x };
    const v4f s1 = { v.y, v.y, v.y, v.y };
    const v4f s2 = { v.z, v.z, v.z, v.z };
    const v4f s3 = { v.w, v.w, v.w, v.w };

#pragma unroll
    for (int r = 0; r < 4; ++r) {
        v4f* row = orow + r * (2048 / 4);   // next output row, same x window
        __builtin_nontemporal_store(s0, row + 0);
        __builtin_nontemporal_store(s1, row + 1);
        __builtin_nontemporal_store(s2, row + 2);
        __builtin_nontemporal_store(s3, row + 3);
    }
}

extern "C" void kernel_launch(void* const* d_in, const int* in_sizes, int n_in,
                              void* d_out, int out_size, void* d_ws, size_t ws_size,
                              hipStream_t stream) {
    (void)n_in; (void)out_size; (void)d_ws; (void)ws_size;

    const float* in  = (const float*)d_in[0];   // [4,3,512,512] f32
    float*       out = (float*)d_out;           // [4,3,2048,2048] f32

    const int n_elems = in_sizes[0];            // 4*3*512*512 = 3,145,728
    const int n_quads = n_elems / 4;            // 786,432 threads

    const int block = 256;                      // 8 wave32s per block
    const int grid  = (n_quads + block - 1) / block;

    UpscalingMethod_35450660061237_kernel<<<grid, block, 0, stream>>>(in, out, n_quads);
}